// SAGE_60799557042640
// MI455X (gfx1250) — compile-verified
//
#include <hip/hip_runtime.h>

#define D 128
#define FANOUT 8
#define N0_DST 124416
#define N1_DST 13824
#define N2_DST 1536

typedef __attribute__((ext_vector_type(2))) float v2f;
typedef __attribute__((ext_vector_type(8))) float v8f;
typedef __attribute__((ext_vector_type(4))) unsigned int u32x4v;
typedef __attribute__((ext_vector_type(8))) int i32x8v;
typedef __attribute__((ext_vector_type(4))) int i32x4v;

__device__ __forceinline__ v8f wmma_f32(v2f a, v2f b, v8f c) {
  // D = A(16x4) * B(4x16) + C, fp32 end to end
  return __builtin_amdgcn_wmma_f32_16x16x4_f32(
      /*neg_a=*/false, a, /*neg_b=*/false, b,
      /*c_mod=*/(short)0, c, /*reuse_a=*/false, /*reuse_b=*/false);
}

#define HAVE_TDM __has_builtin(__builtin_amdgcn_tensor_load_to_lds)

#if HAVE_TDM
// ---------------------------------------------------------------------------
// Tensor Data Mover: DMA a 2D tile (tile_d1 rows x tile_d0 elems, fp32) from
// global (row stride stride_d0 elems) into LDS, inserting pad_amount(+1)
// DWORDs of padding after every 2^(pad_interval+1) DWORDs (hardware does the
// padded-stride LDS layout). Descriptor per CDNA5 ISA.
// This toolchain exposes the 6-arg builtin (g0, g1, g2, g3, g4, cpol); the
// trailing int32x8 group is zero for a plain 2D, non-cluster descriptor.
// Issue from one wave only; tracked on TENSORcnt.
// ---------------------------------------------------------------------------
__device__ __forceinline__ void tdm_load_2d_f32(
    const float* gaddr, void* lds_ptr, unsigned tile_d0, unsigned tile_d1,
    unsigned tensor_d0, unsigned tensor_d1, unsigned stride_d0,
    unsigned pad_interval_code, unsigned pad_amount_code) {
  unsigned long long ga = (unsigned long long)(uintptr_t)gaddr;
  unsigned lds_off = (unsigned)(uintptr_t)lds_ptr;  // low 32 bits = LDS offset
  u32x4v g0;
  g0[0] = 1u;                                    // count=1 (valid), user mode
  g0[1] = lds_off;                               // lds_addr
  g0[2] = (unsigned)(ga & 0xFFFFFFFFu);          // global_addr[31:0]
  g0[3] = (unsigned)((ga >> 32) & 0x1FFFFFFu)    // global_addr[56:32]
          | (2u << 30);                          // type = 2 ("image")
  i32x8v g1;
  g1[0] = (int)((2u << 16)                       // data_size = 4 bytes
                | (1u << 20)                     // pad_enable
                | (pad_interval_code << 22) | (pad_amount_code << 25));
  g1[1] = (int)((tensor_d0 & 0xFFFFu) << 16);    // tensor_dim0[15:0]
  g1[2] = (int)(((tensor_d0 >> 16) & 0xFFFFu) | ((tensor_d1 & 0xFFFFu) << 16));
  g1[3] = (int)(((tensor_d1 >> 16) & 0xFFFFu) | ((tile_d0 & 0xFFFFu) << 16));
  g1[4] = (int)(tile_d1 & 0xFFFFu);              // tile_dim1; tile_dim2 = 0
  g1[5] = (int)stride_d0;                        // tensor_dim0_stride[31:0]
  g1[6] = 0;                                     // stride hi, dim1_stride = 0
  g1[7] = 0;
  i32x4v gz4 = {0, 0, 0, 0};                     // groups 2/3: 2D tensor
  i32x8v gz8 = {0, 0, 0, 0, 0, 0, 0, 0};         // trailing group: unused
  __builtin_amdgcn_tensor_load_to_lds(g0, g1, gz4, gz4, gz8, 0);
}
#endif

// ---------------------------------------------------------------------------
// Gather + mean over FANOUT sampled neighbors. One wave per destination row:
// 32 lanes x float4 = one full 512B row per load -> fully coalesced HBM reads.
// ---------------------------------------------------------------------------
__global__ __launch_bounds__(256) void gather_mean_kernel(
    const float* __restrict__ h, const int* __restrict__ es,
    float* __restrict__ neigh, int n_dst) {
  int wave = (int)((blockIdx.x * blockDim.x + threadIdx.x) >> 5);
  int lane = threadIdx.x & 31;
  if (wave >= n_dst) return;
  const int* e = es + (long)wave * FANOUT;
  float4 acc = make_float4(0.f, 0.f, 0.f, 0.f);
  for (int j = 0; j < FANOUT; ++j) {
    const float4* src = (const float4*)(h + (long)e[j] * D);
    float4 v = src[lane];
    acc.x += v.x; acc.y += v.y; acc.z += v.z; acc.w += v.w;
  }
  const float s = 1.0f / (float)FANOUT;
  acc.x *= s; acc.y *= s; acc.z *= s; acc.w *= s;
  ((float4*)(neigh + (long)wave * D))[lane] = acc;
}

// ---------------------------------------------------------------------------
// out = act( hdst @ Ws + neigh @ Wn + bn )
// Block = 256 threads (8 waves). Block tile: 128 rows x 128 cols.
// Logical K = 256 (stacked [hdst | neigh] vs [Ws ; Wn]), chunked by 32.
// A/W chunks staged into padded LDS by the Tensor Data Mover (wave 0 issues,
// s_wait_tensorcnt 0, barrier). Each wave: 16-row M-slice, 8 N-subtiles of
// f32 16x16x4 WMMA.
// ---------------------------------------------------------------------------
#define A_STRIDE 36
#define W_STRIDE 132

__global__ __launch_bounds__(256) void sage_gemm_kernel(
    const float* __restrict__ hdst, const float* __restrict__ neigh,
    const float* __restrict__ Ws, const float* __restrict__ Wn,
    const float* __restrict__ bn, float* __restrict__ out,
    int n_dst, int do_relu) {
  __shared__ float A_lds[128 * A_STRIDE];
  __shared__ float W_lds[32 * W_STRIDE];

  const int tid = threadIdx.x;
  const int lane = tid & 31;
  const int wave = tid >> 5;       // M-subtile 0..7
  const int row0 = blockIdx.x * 128;
  if (row0 >= n_dst) return;
  const int lm = lane & 15;
  const int hi = lane >> 4;        // 0/1 -> K pair {0,1} vs {2,3}
  const int khalf = hi * 2;

  v8f acc[8] = {};

  for (int kc = 0; kc < 8; ++kc) {           // K = 256, chunks of 32
    const int k0 = kc * 32;
    __syncthreads();
#if HAVE_TDM
    if (tid < 32) {
      // A chunk: 128 rows x 32 K-elems, row stride D, LDS stride 36
      //   pad after every 32 DWORDs (code 4) by 4 DWORDs (code 3)
      const float* asrc = (k0 < 128) ? (hdst + (long)row0 * D + k0)
                                     : (neigh + (long)row0 * D + (k0 - 128));
      tdm_load_2d_f32(asrc, A_lds, /*tile_d0=*/32, /*tile_d1=*/128,
                      /*tensor_d0=*/32, /*tensor_d1=*/128, /*stride=*/D,
                      /*pad_interval=*/4, /*pad_amount=*/3);
      // W chunk: 32 rows(K) x 128 N-elems, row stride D, LDS stride 132
      //   pad after every 128 DWORDs (code 6) by 4 DWORDs (code 3)
      const float* wsrc = (k0 < 128) ? (Ws + (long)k0 * D)
                                     : (Wn + (long)(k0 - 128) * D);
      tdm_load_2d_f32(wsrc, W_lds, /*tile_d0=*/128, /*tile_d1=*/32,
                      /*tensor_d0=*/128, /*tensor_d1=*/32, /*stride=*/D,
                      /*pad_interval=*/6, /*pad_amount=*/3);
      __builtin_amdgcn_s_wait_tensorcnt(0);
    }
#else
    {
      const float* src = (k0 < 128) ? (hdst + (long)row0 * D + k0)
                                    : (neigh + (long)row0 * D + (k0 - 128));
      for (int i = 0; i < 4; ++i) {
        int idx = tid + 256 * i;
        int r = idx >> 3;
        int c4 = idx & 7;
        float4 v = *(const float4*)(src + (long)r * D + c4 * 4);
        float* dst = &A_lds[r * A_STRIDE + c4 * 4];
        dst[0] = v.x; dst[1] = v.y; dst[2] = v.z; dst[3] = v.w;
      }
      const float* wsrc = (k0 < 128) ? (Ws + (long)k0 * D)
                                     : (Wn + (long)(k0 - 128) * D);
      for (int i = 0; i < 4; ++i) {
        int idx = tid + 256 * i;
        int r = idx >> 5;
        int c4 = idx & 31;
        float4 v = *(const float4*)(wsrc + (long)r * D + c4 * 4);
        float* dst = &W_lds[r * W_STRIDE + c4 * 4];
        dst[0] = v.x; dst[1] = v.y; dst[2] = v.z; dst[3] = v.w;
      }
    }
#endif
    __syncthreads();

    // A fragments for this chunk (shared across all 8 N-subtiles)
    v2f afrag[8];
    for (int ks = 0; ks < 8; ++ks) {
      const float* ap = &A_lds[(wave * 16 + lm) * A_STRIDE + ks * 4 + khalf];
      afrag[ks].x = ap[0];
      afrag[ks].y = ap[1];
    }
    for (int nt = 0; nt < 8; ++nt) {
      const int n = nt * 16 + lm;
      for (int ks = 0; ks < 8; ++ks) {
        v2f b;
        b.x = W_lds[(ks * 4 + khalf) * W_STRIDE + n];
        b.y = W_lds[(ks * 4 + khalf + 1) * W_STRIDE + n];
        acc[nt] = wmma_f32(afrag[ks], b, acc[nt]);
      }
    }
  }

  // Epilogue: + bias, optional ReLU, scatter to global
  for (int nt = 0; nt < 8; ++nt) {
    const int col = nt * 16 + lm;
    const float bias = bn[col];
    for (int j = 0; j < 8; ++j) {
      int row = row0 + wave * 16 + j + 8 * hi;
      float v = acc[nt][j] + bias;
      if (do_relu) v = fmaxf(v, 0.0f);
      out[(long)row * D + col] = v;
    }
  }
}

// ---------------------------------------------------------------------------
// Predictor: X = src*pos (rows 0..511) / src*neg (rows 512..1023);
// out = relu(relu(X@pW1+pb1)@pW2+pb2)@pW3 + pb3.  Block = 128 thr (4 waves),
// 64 rows; WMMA for both 128x128 GEMMs, LDS-resident intermediate.
// ---------------------------------------------------------------------------
#define PA_STRIDE 36
#define PW_STRIDE 132
#define PY_STRIDE 132

__global__ __launch_bounds__(128) void predictor_kernel(
    const float* __restrict__ h3,   // 1536 x 128 (src|pos|neg)
    const float* __restrict__ pW1, const float* __restrict__ pb1,
    const float* __restrict__ pW2, const float* __restrict__ pb2,
    const float* __restrict__ pW3, const float* __restrict__ pb3,
    float* __restrict__ out) {      // 1024 floats
  __shared__ float A_lds[64 * PA_STRIDE];
  __shared__ float W_lds[32 * PW_STRIDE];
  __shared__ float Y_lds[64 * PY_STRIDE];

  const int tid = threadIdx.x;
  const int lane = tid & 31;
  const int wave = tid >> 5;  // 0..3
  const int lm = lane & 15;
  const int hi = lane >> 4;
  const int khalf = hi * 2;
  const int row0 = blockIdx.x * 64;

  // ---- GEMM1: relu(X @ pW1 + pb1), X built on the fly ----
  v8f acc[8] = {};
  for (int kc = 0; kc < 4; ++kc) {
    const int k0 = kc * 32;
    __syncthreads();
    for (int i = 0; i < 4; ++i) {           // A chunk: 64 x 32 (elementwise)
      int idx = tid + 128 * i;
      int r = idx >> 3, c4 = idx & 7;
      int g = row0 + r;
      int i0 = g & 511;
      int mate = ((g < 512) ? 512 : 1024) + i0;
      float4 a4 = *(const float4*)(h3 + (long)i0 * D + k0 + c4 * 4);
      float4 b4 = *(const float4*)(h3 + (long)mate * D + k0 + c4 * 4);
      float* dst = &A_lds[r * PA_STRIDE + c4 * 4];
      dst[0] = a4.x * b4.x; dst[1] = a4.y * b4.y;
      dst[2] = a4.z * b4.z; dst[3] = a4.w * b4.w;
    }
#if HAVE_TDM
    if (tid < 32) {
      tdm_load_2d_f32(pW1 + (long)k0 * D, W_lds, 128, 32, 128, 32, D, 6, 3);
      __builtin_amdgcn_s_wait_tensorcnt(0);
    }
#else
    for (int i = 0; i < 8; ++i) {           // W chunk: 32 x 128
      int idx = tid + 128 * i;
      int r = idx >> 5, c4 = idx & 31;
      float4 v = *(const float4*)(pW1 + (long)(k0 + r) * D + c4 * 4);
      float* dst = &W_lds[r * PW_STRIDE + c4 * 4];
      dst[0] = v.x; dst[1] = v.y; dst[2] = v.z; dst[3] = v.w;
    }
#endif
    __syncthreads();
    v2f afrag[8];
    for (int ks = 0; ks < 8; ++ks) {
      const float* ap = &A_lds[(wave * 16 + lm) * PA_STRIDE + ks * 4 + khalf];
      afrag[ks].x = ap[0]; afrag[ks].y = ap[1];
    }
    for (int nt = 0; nt < 8; ++nt) {
      int n = nt * 16 + lm;
      for (int ks = 0; ks < 8; ++ks) {
        v2f b;
        b.x = W_lds[(ks * 4 + khalf) * PW_STRIDE + n];
        b.y = W_lds[(ks * 4 + khalf + 1) * PW_STRIDE + n];
        acc[nt] = wmma_f32(afrag[ks], b, acc[nt]);
      }
    }
  }
  __syncthreads();
  for (int nt = 0; nt < 8; ++nt) {
    int col = nt * 16 + lm;
    float bias = pb1[col];
    for (int j = 0; j < 8; ++j) {
      int r = wave * 16 + j + 8 * hi;
      Y_lds[r * PY_STRIDE + col] = fmaxf(acc[nt][j] + bias, 0.0f);
    }
  }
  __syncthreads();

  // ---- GEMM2: relu(Y @ pW2 + pb2) ----
  v8f acc2[8] = {};
  for (int kc = 0; kc < 4; ++kc) {
    const int k0 = kc * 32;
    __syncthreads();
#if HAVE_TDM
    if (tid < 32) {
      tdm_load_2d_f32(pW2 + (long)k0 * D, W_lds, 128, 32, 128, 32, D, 6, 3);
      __builtin_amdgcn_s_wait_tensorcnt(0);
    }
#else
    for (int i = 0; i < 8; ++i) {
      int idx = tid + 128 * i;
      int r = idx >> 5, c4 = idx & 31;
      float4 v = *(const float4*)(pW2 + (long)(k0 + r) * D + c4 * 4);
      float* dst = &W_lds[r * PW_STRIDE + c4 * 4];
      dst[0] = v.x; dst[1] = v.y; dst[2] = v.z; dst[3] = v.w;
    }
#endif
    __syncthreads();
    v2f afrag[8];
    for (int ks = 0; ks < 8; ++ks) {
      const float* ap = &Y_lds[(wave * 16 + lm) * PY_STRIDE + k0 + ks * 4 + khalf];
      afrag[ks].x = ap[0]; afrag[ks].y = ap[1];
    }
    for (int nt = 0; nt < 8; ++nt) {
      int n = nt * 16 + lm;
      for (int ks = 0; ks < 8; ++ks) {
        v2f b;
        b.x = W_lds[(ks * 4 + khalf) * PW_STRIDE + n];
        b.y = W_lds[(ks * 4 + khalf + 1) * PW_STRIDE + n];
        acc2[nt] = wmma_f32(afrag[ks], b, acc2[nt]);
      }
    }
  }
  __syncthreads();
  for (int nt = 0; nt < 8; ++nt) {
    int col = nt * 16 + lm;
    float bias = pb2[col];
    for (int j = 0; j < 8; ++j) {
      int r = wave * 16 + j + 8 * hi;
      Y_lds[r * PY_STRIDE + col] = fmaxf(acc2[nt][j] + bias, 0.0f);
    }
  }
  __syncthreads();

  // ---- out = Y2 @ pW3 + pb3 (128-dot per row) ----
  if (tid < 64) {
    float s = pb3[0];
    for (int k = 0; k < D; ++k) s += Y_lds[tid * PY_STRIDE + k] * pW3[k];
    out[row0 + tid] = s;
  }
}

// ---------------------------------------------------------------------------
extern "C" void kernel_launch(void* const* d_in, const int* in_sizes, int n_in,
                              void* d_out, int out_size, void* d_ws,
                              size_t ws_size, hipStream_t stream) {
  const float* h   = (const float*)d_in[0];
  const int* es0   = (const int*)d_in[1];
  const int* es1   = (const int*)d_in[2];
  const int* es2   = (const int*)d_in[3];
  const float* Ws0 = (const float*)d_in[4];
  const float* Wn0 = (const float*)d_in[5];
  const float* bn0 = (const float*)d_in[6];
  const float* Ws1 = (const float*)d_in[7];
  const float* Wn1 = (const float*)d_in[8];
  const float* bn1 = (const float*)d_in[9];
  const float* Ws2 = (const float*)d_in[10];
  const float* Wn2 = (const float*)d_in[11];
  const float* bn2 = (const float*)d_in[12];
  const float* pW1 = (const float*)d_in[13];
  const float* pb1 = (const float*)d_in[14];
  const float* pW2 = (const float*)d_in[15];
  const float* pb2 = (const float*)d_in[16];
  const float* pW3 = (const float*)d_in[17];
  const float* pb3 = (const float*)d_in[18];

  float* ws = (float*)d_ws;
  float* neigh = ws;                              // N0_DST*128 (reused each layer)
  float* h1 = neigh + (size_t)N0_DST * D;         // N0_DST*128
  float* h2 = h1 + (size_t)N0_DST * D;            // N1_DST*128
  float* h3 = h2 + (size_t)N1_DST * D;            // N2_DST*128

  // Layer 0
  gather_mean_kernel<<<N0_DST / 8, 256, 0, stream>>>(h, es0, neigh, N0_DST);
  sage_gemm_kernel<<<N0_DST / 128, 256, 0, stream>>>(h, neigh, Ws0, Wn0, bn0,
                                                     h1, N0_DST, 1);
  // Layer 1
  gather_mean_kernel<<<N1_DST / 8, 256, 0, stream>>>(h1, es1, neigh, N1_DST);
  sage_gemm_kernel<<<N1_DST / 128, 256, 0, stream>>>(h1, neigh, Ws1, Wn1, bn1,
                                                     h2, N1_DST, 1);
  // Layer 2 (no ReLU)
  gather_mean_kernel<<<N2_DST / 8, 256, 0, stream>>>(h2, es2, neigh, N2_DST);
  sage_gemm_kernel<<<N2_DST / 128, 256, 0, stream>>>(h2, neigh, Ws2, Wn2, bn2,
                                                     h3, N2_DST, 0);
  // Predictor
  predictor_kernel<<<1024 / 64, 128, 0, stream>>>(h3, pW1, pb1, pW2, pb2, pW3,
                                                  pb3, (float*)d_out);
}